// BV_Model_56298431316395
// MI455X (gfx1250) — compile-verified
//
#include <hip/hip_runtime.h>
#include <hip/hip_bf16.h>

// ---------------------------------------------------------------------------
// BV_Model (GENConv-style quantized GNN) on MI455X (gfx1250, wave32).
//
// Memory-bound edge aggregation (fp32 atomics, recompute edge encoder on the
// fly to cut HBM traffic), WMMA f32 16x16x4 for the per-layer MLP GEMMs
// (exact fp32, matches the fp32 reference).
// ---------------------------------------------------------------------------

typedef float v2f __attribute__((ext_vector_type(2)));
typedef float v8f __attribute__((ext_vector_type(8)));

#define CQ 32          // channel count C
#define BN_EPS 1e-5f
#define GEN_EPS 1e-7f

// ap_fixed<16,6> fake quantization: step 2^-10, clip [-32, 32-step].
// jnp.round is round-half-to-even -> rintf (v_rndne_f32).
__device__ __forceinline__ float qfix(float x) {
    float y = rintf(x * 1024.0f) * 0.0009765625f;
    return fminf(fmaxf(y, -32.0f), 32.0f - 0.0009765625f);
}

// float atomic max via sign-split int/uint trick (init with -inf works).
__device__ __forceinline__ void atomicMaxF(float* addr, float val) {
    if (val >= 0.0f)
        atomicMax((int*)addr, __float_as_int(val));
    else
        atomicMin((unsigned int*)addr, __float_as_uint(val));
}

// ---------------------------------------------------------------------------
// init per-layer segment buffers: mx=-inf, den=0, num=0
// ---------------------------------------------------------------------------
__global__ void k_init_layer(float* __restrict__ mx, float* __restrict__ den,
                             float* __restrict__ num, int n) {
    int i = blockIdx.x * blockDim.x + threadIdx.x;
    if (i < n) {
        mx[i]  = -__builtin_inff();
        den[i] = 0.0f;
        num[i] = 0.0f;
    }
}

__global__ void k_zero(float* __restrict__ p, int n) {
    int i = blockIdx.x * blockDim.x + threadIdx.x;
    if (i < n) p[i] = 0.0f;
}

// ---------------------------------------------------------------------------
// node encoder: h = bn(qlin(x[N,8] @ W_node[8,32] + b_node))
// one lane per (node, channel); 8 nodes per 256-thread block
// ---------------------------------------------------------------------------
__global__ void k_encode_nodes(const float* __restrict__ x,
                               const float* __restrict__ Wn,
                               const float* __restrict__ bn_,
                               const float* __restrict__ g,
                               const float* __restrict__ b,
                               const float* __restrict__ m,
                               const float* __restrict__ v,
                               float* __restrict__ h, int N) {
    int t = blockIdx.x * blockDim.x + threadIdx.x;
    int node = t >> 5, c = t & 31;
    if (node >= N) return;
    float acc = 0.0f;
#pragma unroll
    for (int k = 0; k < 8; ++k)
        acc += qfix(x[node * 8 + k]) * qfix(Wn[k * CQ + c]);
    float z = qfix(acc + qfix(bn_[c]));
    z = (z - m[c]) * rsqrtf(v[c] + BN_EPS) * g[c] + b[c];
    h[node * CQ + c] = z;
}

// edge feature channel c, recomputed on the fly (16B/edge instead of 128B)
__device__ __forceinline__ float edge_feat(const float* __restrict__ ea, int e,
                                           const float* __restrict__ We,
                                           const float* __restrict__ be,
                                           const float* __restrict__ bg,
                                           const float* __restrict__ bb,
                                           const float* __restrict__ bm,
                                           const float* __restrict__ bv, int c) {
    float acc = 0.0f;
#pragma unroll
    for (int k = 0; k < 4; ++k)
        acc += qfix(ea[e * 4 + k]) * qfix(We[k * CQ + c]);   // ea addr uniform per wave
    float ec = qfix(acc + qfix(be[c]));
    return (ec - bm[c]) * rsqrtf(bv[c] + BN_EPS) * bg[c] + bb[c];
}

// ---------------------------------------------------------------------------
// edge pass A: segment max of s = t[l] * (relu(h[src]+e)+eps) over dst
// ---------------------------------------------------------------------------
__global__ void k_edge_max(const float* __restrict__ ea,
                           const float* __restrict__ We, const float* __restrict__ be,
                           const float* __restrict__ bg, const float* __restrict__ bb,
                           const float* __restrict__ bm, const float* __restrict__ bv,
                           const float* __restrict__ h, const int* __restrict__ ei,
                           const float* __restrict__ tptr, int layer,
                           float* __restrict__ mx, int E) {
    int t = blockIdx.x * blockDim.x + threadIdx.x;
    int e = t >> 5, c = t & 31;
    if (e >= E) return;
    int src = ei[e];           // wave-uniform loads
    int dst = ei[E + e];
    float ec = edge_feat(ea, e, We, be, bg, bb, bm, bv, c);
    float m = fmaxf(h[src * CQ + c] + ec, 0.0f) + GEN_EPS;   // coalesced 128B gather
    atomicMaxF(&mx[dst * CQ + c], tptr[layer] * m);
}

// ---------------------------------------------------------------------------
// edge pass B: ex = exp(s - mx[dst]); den += ex; num += ex*m
// (nodes with no in-edges are never read here, so no isfinite fixup needed)
// ---------------------------------------------------------------------------
__global__ void k_edge_sum(const float* __restrict__ ea,
                           const float* __restrict__ We, const float* __restrict__ be,
                           const float* __restrict__ bg, const float* __restrict__ bb,
                           const float* __restrict__ bm, const float* __restrict__ bv,
                           const float* __restrict__ h, const int* __restrict__ ei,
                           const float* __restrict__ tptr, int layer,
                           const float* __restrict__ mx,
                           float* __restrict__ den, float* __restrict__ num, int E) {
    int t = blockIdx.x * blockDim.x + threadIdx.x;
    int e = t >> 5, c = t & 31;
    if (e >= E) return;
    int src = ei[e];
    int dst = ei[E + e];
    float ec = edge_feat(ea, e, We, be, bg, bb, bm, bv, c);
    float m = fmaxf(h[src * CQ + c] + ec, 0.0f) + GEN_EPS;
    float s = tptr[layer] * m;
    float ex = __expf(s - mx[dst * CQ + c]);
    atomicAdd(&den[dst * CQ + c], ex);
    atomicAdd(&num[dst * CQ + c], ex * m);
}

// ---------------------------------------------------------------------------
// per-layer node update (fused):
//   h2 = h + num/max(den,1e-16)
//   z  = qlin(h2, W1[l], b1[l]); z = bn1(z); z = q(relu(z))
//   z  = qlin(z, W2[l], b2[l]);  h = z + h
// 16 nodes per block, 128 threads (4 waves). GEMMs via V_WMMA_F32_16X16X4_F32:
//   A 16x4 f32 frag: lanes 0-15 hold {K0,K1}, lanes 16-31 hold {K2,K3} (ISA 7.12.2)
//   B 4x16 f32 frag mirrors A (rows split across lane halves)
//   C/D 16x16: VGPR i = row i (lanes 0-15) / row i+8 (lanes 16-31), N = lane%16
// ---------------------------------------------------------------------------
__global__ void __launch_bounds__(128)
k_node_update(float* __restrict__ h,
              const float* __restrict__ den, const float* __restrict__ num,
              const float* __restrict__ W1, const float* __restrict__ b1,
              const float* __restrict__ g1, const float* __restrict__ be1,
              const float* __restrict__ m1, const float* __restrict__ v1,
              const float* __restrict__ W2, const float* __restrict__ b2,
              int layer, int N) {
    __shared__ float sA[16 * 33];    // q(h2) tile, padded stride
    __shared__ float sW1[32 * 65];   // q(W1[l]) 32x64, padded
    __shared__ float sZ[16 * 65];    // q(relu(bn(z1))) 16x64, padded
    __shared__ float sW2[64 * 33];   // q(W2[l]) 64x32, padded

    const int tid  = threadIdx.x;
    const int wave = tid >> 5;
    const int lane = tid & 31;
    const int lh   = lane >> 4;      // lane half (0/1)
    const int lm   = lane & 15;
    const int nb   = blockIdx.x * 16;

    const float* W1l = W1 + (size_t)layer * 32 * 64;
    const float* W2l = W2 + (size_t)layer * 64 * 32;

    for (int i = tid; i < 32 * 64; i += 128) {
        int r = i >> 6, cc = i & 63;
        sW1[r * 65 + cc] = qfix(W1l[i]);
    }
    for (int i = tid; i < 64 * 32; i += 128) {
        int r = i >> 5, cc = i & 31;
        sW2[r * 33 + cc] = qfix(W2l[i]);
    }
    for (int i = tid; i < 16 * CQ; i += 128) {
        int r = i >> 5, cc = i & 31;
        int node = min(nb + r, N - 1);                 // clamp; keep EXEC full
        float d  = den[node * CQ + cc];
        float a  = num[node * CQ + cc] / fmaxf(d, 1e-16f);
        sA[r * 33 + cc] = qfix(h[node * CQ + cc] + a); // input quant of qlin #1
    }
    __syncthreads();

    // ---- GEMM1: [16,32] @ [32,64]; wave w covers output cols w*16..w*16+15
    {
        const int col = wave * 16 + lm;
        v8f acc = {};
#pragma unroll
        for (int kk = 0; kk < 32; kk += 4) {
            const int k0 = kk + 2 * lh;
            v2f a, b;
            a.x = sA[lm * 33 + k0];
            a.y = sA[lm * 33 + k0 + 1];
            b.x = sW1[k0 * 65 + col];
            b.y = sW1[(k0 + 1) * 65 + col];
            acc = __builtin_amdgcn_wmma_f32_16x16x4_f32(
                false, a, false, b, (short)0, acc, false, false);
        }
        const int gcol = layer * 64 + col;
        const float qb = qfix(b1[gcol]);
        const float rs = rsqrtf(v1[gcol] + BN_EPS) * g1[gcol];
        const float mm = m1[gcol];
        const float bb = be1[gcol];
#pragma unroll
        for (int i = 0; i < 8; ++i) {
            int row = i + 8 * lh;
            float z = qfix(acc[i] + qb);               // output quant of qlin #1
            z = (z - mm) * rs + bb;                    // bn1
            z = qfix(fmaxf(z, 0.0f));                  // quantized relu
            sZ[row * 65 + col] = z;
        }
    }
    __syncthreads();

    // ---- GEMM2: [16,64] @ [64,32]; waves 0,1 cover output cols
    if (wave < 2) {
        const int col = wave * 16 + lm;
        v8f acc = {};
#pragma unroll
        for (int kk = 0; kk < 64; kk += 4) {
            const int k0 = kk + 2 * lh;
            v2f a, b;
            a.x = sZ[lm * 65 + k0];
            a.y = sZ[lm * 65 + k0 + 1];
            b.x = sW2[k0 * 33 + col];
            b.y = sW2[(k0 + 1) * 33 + col];
            acc = __builtin_amdgcn_wmma_f32_16x16x4_f32(
                false, a, false, b, (short)0, acc, false, false);
        }
        const float qb = qfix(b2[layer * CQ + col]);
#pragma unroll
        for (int i = 0; i < 8; ++i) {
            int row  = i + 8 * lh;
            int node = nb + row;
            if (node < N) {
                float z = qfix(acc[i] + qb);
                h[node * CQ + col] = z + h[node * CQ + col];  // residual (read-then-write, same thread)
            }
        }
    }
}

// ---------------------------------------------------------------------------
// global mean pool (atomics) + quantized head with sigmoid
// ---------------------------------------------------------------------------
__global__ void k_pool(const float* __restrict__ h, const int* __restrict__ batch,
                       float* __restrict__ pooled, float* __restrict__ cnt, int N) {
    int t = blockIdx.x * blockDim.x + threadIdx.x;
    int node = t >> 5, c = t & 31;
    if (node >= N) return;
    int g = batch[node];
    atomicAdd(&pooled[g * CQ + c], h[node * CQ + c]);
    if (c == 0) atomicAdd(&cnt[g], 1.0f);
}

__global__ void k_head(const float* __restrict__ pooled, const float* __restrict__ cnt,
                       const float* __restrict__ Wo, const float* __restrict__ bo,
                       float* __restrict__ out, int G) {
    int g = blockIdx.x * blockDim.x + threadIdx.x;
    if (g >= G) return;
    float inv = 1.0f / fmaxf(cnt[g], 1.0f);
    float acc = 0.0f;
#pragma unroll
    for (int c = 0; c < CQ; ++c)
        acc += qfix(pooled[g * CQ + c] * inv) * qfix(Wo[c]);
    float z = qfix(acc + qfix(bo[0]));
    z = 1.0f / (1.0f + __expf(-z));
    out[g] = qfix(z);
}

// ---------------------------------------------------------------------------
extern "C" void kernel_launch(void* const* d_in, const int* in_sizes, int n_in,
                              void* d_out, int out_size, void* d_ws, size_t ws_size,
                              hipStream_t stream) {
    const float* x      = (const float*)d_in[0];
    const float* ea     = (const float*)d_in[1];
    const float* W_node = (const float*)d_in[2];
    const float* b_node = (const float*)d_in[3];
    const float* W_edge = (const float*)d_in[4];
    const float* b_edge = (const float*)d_in[5];
    const float* bnn_g  = (const float*)d_in[6];
    const float* bnn_b  = (const float*)d_in[7];
    const float* bnn_m  = (const float*)d_in[8];
    const float* bnn_v  = (const float*)d_in[9];
    const float* bne_g  = (const float*)d_in[10];
    const float* bne_b  = (const float*)d_in[11];
    const float* bne_m  = (const float*)d_in[12];
    const float* bne_v  = (const float*)d_in[13];
    const float* t      = (const float*)d_in[14];
    const float* W1     = (const float*)d_in[15];
    const float* b1     = (const float*)d_in[16];
    const float* bn1_g  = (const float*)d_in[17];
    const float* bn1_b  = (const float*)d_in[18];
    const float* bn1_m  = (const float*)d_in[19];
    const float* bn1_v  = (const float*)d_in[20];
    const float* W2     = (const float*)d_in[21];
    const float* b2     = (const float*)d_in[22];
    const float* W_out  = (const float*)d_in[23];
    const float* b_out  = (const float*)d_in[24];
    const int*   ei     = (const int*)d_in[25];
    const int*   batch  = (const int*)d_in[26];

    const int N = in_sizes[0] / 8;
    const int E = in_sizes[1] / 4;
    const int L = in_sizes[14];
    const int G = out_size;          // output is [G,1] fp32

    // workspace layout (fp32): h | mx | den | num | pooled | cnt
    float* ws     = (float*)d_ws;
    float* h      = ws;
    float* mx     = h   + (size_t)N * CQ;
    float* den    = mx  + (size_t)N * CQ;
    float* num    = den + (size_t)N * CQ;
    float* pooled = num + (size_t)N * CQ;
    float* cnt    = pooled + (size_t)G * CQ;

    const int nElemN  = N * CQ;
    const int nBlkN   = (nElemN + 255) / 256;                 // (node,chan) grids
    const long eElems = (long)E * CQ;
    const int nBlkE   = (int)((eElems + 255) / 256);          // (edge,chan) grids
    const int nBlkU   = (N + 15) / 16;                        // 16-node MLP tiles

    k_encode_nodes<<<nBlkN, 256, 0, stream>>>(x, W_node, b_node, bnn_g, bnn_b,
                                              bnn_m, bnn_v, h, N);

    for (int l = 0; l < L; ++l) {
        k_init_layer<<<nBlkN, 256, 0, stream>>>(mx, den, num, nElemN);
        k_edge_max<<<nBlkE, 256, 0, stream>>>(ea, W_edge, b_edge, bne_g, bne_b,
                                              bne_m, bne_v, h, ei, t, l, mx, E);
        k_edge_sum<<<nBlkE, 256, 0, stream>>>(ea, W_edge, b_edge, bne_g, bne_b,
                                              bne_m, bne_v, h, ei, t, l, mx,
                                              den, num, E);
        k_node_update<<<nBlkU, 128, 0, stream>>>(h, den, num, W1, b1, bn1_g,
                                                 bn1_b, bn1_m, bn1_v, W2, b2,
                                                 l, N);
    }

    k_zero<<<(G * (CQ + 1) + 255) / 256, 256, 0, stream>>>(pooled, G * (CQ + 1));
    k_pool<<<nBlkN, 256, 0, stream>>>(h, batch, pooled, cnt, N);
    k_head<<<(G + 255) / 256, 256, 0, stream>>>(pooled, cnt, W_out, b_out,
                                                (float*)d_out, G);
}